// SelfConnectionArea_27358941675618
// MI455X (gfx1250) — compile-verified
//
#include <hip/hip_runtime.h>

// SNN forward + STDP rank-1 updates for MI455X (gfx1250, wave32).
// Everything is HBM-bound (~1.07 GB traffic/call); GEMV phase uses
// V_WMMA_F32_16X16X4_F32 so the weight stream feeds the matrix pipe.

#define N       8192
#define KSPLIT  8
#define KSLICE  (N / KSPLIT)   // 1024 rows of W per block.y slice
#define V_TH    0.5f
#define DECAY   0.99f

typedef float v2f __attribute__((ext_vector_type(2)));
typedef float v8f __attribute__((ext_vector_type(8)));

// ---------------------------------------------------------------------------
// Phase 1: partial[ks][n] = sum_{k in slice ks} x[k]*w1[k][n] + x1[k]*w2[k][n]
// One wave per 16-column tile. M=1 GEMV mapped onto 16x16x4 f32 WMMA:
//   A (16x4): row M=0 carries the vector (lane0 -> K0,K1; lane16 -> K2,K3),
//             all other rows zero, so D rows 1..15 stay zero.
//   B (4x16): lanes 0-15 hold K0(V0)/K1(V1), lanes 16-31 hold K2(V0)/K3(V1),
//             N = lane & 15.
//   D row M=0 lives in acc[0] of lanes 0..15.
// ---------------------------------------------------------------------------
__device__ inline v8f gemv_slice_wmma(const float* __restrict__ vec,
                                      const float* __restrict__ Wcol, // W + n
                                      int k0, int lane, int koff, v8f acc) {
  for (int k = k0; k < k0 + KSLICE; k += 4) {
    float4 xv = *(const float4*)(vec + k);          // broadcast load, all lanes
    v2f a, b;
    a.x = (lane == 0) ? xv.x : ((lane == 16) ? xv.z : 0.0f);
    a.y = (lane == 0) ? xv.y : ((lane == 16) ? xv.w : 0.0f);
    b.x = Wcol[(k + koff) * N];                     // coalesced across lane&15
    b.y = Wcol[(k + koff + 1) * N];
    acc = __builtin_amdgcn_wmma_f32_16x16x4_f32(
        /*neg_a=*/false, a, /*neg_b=*/false, b,
        /*c_mod=*/(short)0, acc, /*reuse_a=*/false, /*reuse_b=*/false);
  }
  return acc;
}

__global__ __launch_bounds__(128)
void snn_gemv_wmma(const float* __restrict__ x,  const float* __restrict__ x1,
                   const float* __restrict__ w1, const float* __restrict__ w2,
                   float* __restrict__ partial) {
  const int wave = threadIdx.x >> 5;
  const int lane = threadIdx.x & 31;
  const int n    = blockIdx.x * 64 + wave * 16 + (lane & 15);
  const int koff = (lane & 16) >> 3;   // 0 for lanes 0-15, 2 for lanes 16-31
  const int k0   = blockIdx.y * KSLICE;

  v8f acc = {};
  acc = gemv_slice_wmma(x,  w1 + n, k0, lane, koff, acc);
  acc = gemv_slice_wmma(x1, w2 + n, k0, lane, koff, acc);

  if (lane < 16)                        // D row M=0: VGPR0, lanes 0..15
    partial[blockIdx.y * N + n] = acc[0];
}

// ---------------------------------------------------------------------------
// Phase 2: reduce K-slices, IF-neuron threshold, STDP traces.
// ---------------------------------------------------------------------------
__global__ __launch_bounds__(256)
void snn_spike_trace(const float* __restrict__ partial,
                     const float* __restrict__ mem,
                     const float* __restrict__ x,
                     const float* __restrict__ x1,
                     const float* __restrict__ trace1,
                     const float* __restrict__ trace2,
                     float* __restrict__ s_out,
                     float* __restrict__ t1,
                     float* __restrict__ t2) {
  const int n = blockIdx.x * blockDim.x + threadIdx.x;
  if (n >= N) return;
  float isum = 0.0f;
#pragma unroll
  for (int ks = 0; ks < KSPLIT; ++ks) isum += partial[ks * N + n];
  const float m_new = mem[n] + isum;
  s_out[n] = (m_new > V_TH) ? 1.0f : 0.0f;
  t1[n] = trace1[n] * DECAY + x[n];
  t2[n] = trace2[n] * DECAY + x1[n];
}

// ---------------------------------------------------------------------------
// Phase 3: dw1 = t1^T @ s, dw2 = t2^T @ s. Store-bandwidth bound (512 MB);
// two float4 (global_store_b128) stores per thread, fully coalesced.
// ---------------------------------------------------------------------------
__global__ __launch_bounds__(256)
void snn_outer(const float* __restrict__ t1, const float* __restrict__ t2,
               const float* __restrict__ s,
               float* __restrict__ dw1, float* __restrict__ dw2) {
  const int    n4  = blockIdx.x * blockDim.x + threadIdx.x; // 0..2047 chunks
  const int    i   = blockIdx.y;
  const float4 s4  = ((const float4*)s)[n4];                // hot, L2-resident
  const float  a1  = t1[i];
  const float  a2  = t2[i];
  const size_t row = (size_t)i * (N / 4);
  float4 r1 = make_float4(a1 * s4.x, a1 * s4.y, a1 * s4.z, a1 * s4.w);
  float4 r2 = make_float4(a2 * s4.x, a2 * s4.y, a2 * s4.z, a2 * s4.w);
  ((float4*)dw1)[row + n4] = r1;
  ((float4*)dw2)[row + n4] = r2;
}

// ---------------------------------------------------------------------------
extern "C" void kernel_launch(void* const* d_in, const int* in_sizes, int n_in,
                              void* d_out, int out_size, void* d_ws, size_t ws_size,
                              hipStream_t stream) {
  const float* x      = (const float*)d_in[0];
  const float* x1     = (const float*)d_in[1];
  const float* mem    = (const float*)d_in[2];
  const float* trace1 = (const float*)d_in[3];
  const float* trace2 = (const float*)d_in[4];
  const float* w1     = (const float*)d_in[5];
  const float* w2     = (const float*)d_in[6];

  float* out_s = (float*)d_out;                 // [N] spikes
  float* dw1   = out_s + N;                     // [N*N]
  float* dw2   = dw1 + (size_t)N * N;           // [N*N]

  float* partial = (float*)d_ws;                // [KSPLIT*N], fully rewritten
  float* t1      = partial + KSPLIT * N;        // [N]
  float* t2      = t1 + N;                      // [N]

  snn_gemv_wmma<<<dim3(N / 64, KSPLIT), 128, 0, stream>>>(x, x1, w1, w2, partial);
  snn_spike_trace<<<dim3(N / 256), 256, 0, stream>>>(partial, mem, x, x1,
                                                     trace1, trace2, out_s, t1, t2);
  snn_outer<<<dim3(N / 1024, N), 256, 0, stream>>>(t1, t2, out_s, dw1, dw2);
}